// SynapsePredictor_11828339933535
// MI455X (gfx1250) — compile-verified
//
#include <hip/hip_runtime.h>
#include <hip/hip_bf16.h>

typedef __attribute__((ext_vector_type(2))) float v2f;
typedef __attribute__((ext_vector_type(8))) float v8f;

// D = A(16x4 f32) * B(4x16 f32) + C, per CDNA5 V_WMMA_F32_16X16X4_F32
__device__ __forceinline__ v8f wmma_k4(v2f a, v2f b, v8f c) {
    return __builtin_amdgcn_wmma_f32_16x16x4_f32(
        /*neg_a=*/false, a, /*neg_b=*/false, b,
        /*c_mod=*/(short)0, c, /*reuse_a=*/false, /*reuse_b=*/false);
}

#define FEAT 512

__global__ void zero_kernel(float* __restrict__ p, int n) {
    int i = blockIdx.x * blockDim.x + threadIdx.x;
    if (i < n) p[i] = 0.0f;
}

__global__ void count_kernel(const int* __restrict__ dst, float* __restrict__ cnt, int E) {
    int e = blockIdx.x * blockDim.x + threadIdx.x;
    if (e < E) atomicAdd(&cnt[dst[e]], 1.0f);
}

// agg[dst[e], :] += x[src[e], :] * ew[e]   (128 threads per edge, float4 each)
__global__ void scatter_kernel(const float* __restrict__ x, const int* __restrict__ src,
                               const int* __restrict__ dst, const float* __restrict__ ew,
                               float* __restrict__ agg, int E) {
    long long t = (long long)blockIdx.x * blockDim.x + threadIdx.x;
    int e = (int)(t >> 7);
    if (e >= E) return;
    int f = ((int)t & 127) << 2;
    float w = ew[e];
    const float4 v = *(const float4*)(x + (size_t)src[e] * FEAT + f);
    float* o = agg + (size_t)dst[e] * FEAT + f;
    atomicAdd(o + 0, v.x * w);
    atomicAdd(o + 1, v.y * w);
    atomicAdd(o + 2, v.z * w);
    atomicAdd(o + 3, v.w * w);
}

__global__ void normalize_kernel(float* __restrict__ agg, const float* __restrict__ cnt, int n) {
    int i = blockIdx.x * blockDim.x + threadIdx.x;
    if (i >= n) return;
    int row = i >> 9; // FEAT == 512
    agg[i] *= 1.0f / fmaxf(cnt[row], 1.0f);
}

// C[M,512] = A1 @ W1 + A2 @ W2 + bias (+ relu).
// One 16(M) x 64(N) tile per wave: the A fragment is loaded once per k-step
// and reused by 4 WMMAs (4 accumulators), cutting VMEM traffic ~1.6x vs a
// 16x16 tile.  M must be a multiple of 16 (10000 = 625*16).
__global__ __launch_bounds__(256)
void node_gemm_dual(const float* __restrict__ A1, const float* __restrict__ W1,
                    const float* __restrict__ A2, const float* __restrict__ W2,
                    const float* __restrict__ bias, float* __restrict__ C,
                    int M, int doRelu) {
    const int K = FEAT, N = FEAT;
    int wave = threadIdx.x >> 5;
    int lane = threadIdx.x & 31;
    int tile = blockIdx.x * 8 + wave;
    const int ngrp = N / 64;              // 8 n-groups of 64 columns
    int mt = tile / ngrp, ng = tile % ngrp;
    if (mt * 16 >= M) return;

    int mrow = lane & 15;
    int n0 = ng * 64 + mrow;
    int koff = (lane < 16) ? 0 : 2;       // upper half-wave holds K=2,3

    v8f acc[4] = {{}, {}, {}, {}};
    const float* a1row = A1 + (size_t)(mt * 16 + mrow) * K;
    const float* a2row = A2 + (size_t)(mt * 16 + mrow) * K;

    for (int k = 0; k < K; k += 4) {
        v2f a = *(const v2f*)(a1row + k + koff);
        const float* w0 = W1 + (size_t)(k + koff) * N + n0;
        __builtin_prefetch(w0 + 4 * N, 0, 1);   // next k-step's B rows
#pragma unroll
        for (int t = 0; t < 4; ++t) {
            v2f b;
            b.x = w0[t * 16];
            b.y = w0[N + t * 16];
            acc[t] = wmma_k4(a, b, acc[t]);
        }
    }
    for (int k = 0; k < K; k += 4) {
        v2f a = *(const v2f*)(a2row + k + koff);
        const float* w0 = W2 + (size_t)(k + koff) * N + n0;
        __builtin_prefetch(w0 + 4 * N, 0, 1);
#pragma unroll
        for (int t = 0; t < 4; ++t) {
            v2f b;
            b.x = w0[t * 16];
            b.y = w0[N + t * 16];
            acc[t] = wmma_k4(a, b, acc[t]);
        }
    }

    int mbase = mt * 16 + ((lane < 16) ? 0 : 8);
#pragma unroll
    for (int t = 0; t < 4; ++t) {
        float bv = bias[n0 + t * 16];
#pragma unroll
        for (int j = 0; j < 8; ++j) {
            float v = acc[t][j] + bv;
            if (doRelu) v = fmaxf(v, 0.0f);
            C[(size_t)(mbase + j) * N + n0 + t * 16] = v;
        }
    }
}

// Decoder: per wave, 16 label-edges.  h1(16x64) = relu([z[s]|z[d]|ew] @ Wd1 + bd1)
// computed as 4 WMMA tiles sharing the A fragment, then 64->1 via LDS reduction
// (fuses away the 41 MB h1 buffer and its round-trip).
__global__ __launch_bounds__(256)
void decode_kernel(const float* __restrict__ z, const int* __restrict__ eli,
                   const float* __restrict__ expw,
                   const float* __restrict__ Wd1, const float* __restrict__ bd1,
                   const float* __restrict__ Wd2, const float* __restrict__ bd2,
                   float* __restrict__ out, int E) {
    const int H = FEAT;
    __shared__ float lds[8 * 16 * 64]; // 32 KB

    int wave = threadIdx.x >> 5;
    int lane = threadIdx.x & 31;
    int base = blockIdx.x * 128 + wave * 16;
    int mrow = lane & 15;
    int r = base + mrow;
    int sidx = eli[r];
    int didx = eli[E + r];
    int koff = (lane < 16) ? 0 : 2;

    v8f acc[4] = {{}, {}, {}, {}};
    const float* se = z + (size_t)sidx * H;
    const float* de = z + (size_t)didx * H;

    for (int k = 0; k < H; k += 4) {
        v2f a = *(const v2f*)(se + k + koff);
        const float* w0 = Wd1 + (size_t)(k + koff) * 64 + mrow;
#pragma unroll
        for (int t = 0; t < 4; ++t) {
            v2f b;
            b.x = w0[t * 16];
            b.y = w0[64 + t * 16];
            acc[t] = wmma_k4(a, b, acc[t]);
        }
    }
    for (int k = 0; k < H; k += 4) {
        v2f a = *(const v2f*)(de + k + koff);
        const float* w0 = Wd1 + (size_t)(H + k + koff) * 64 + mrow;
#pragma unroll
        for (int t = 0; t < 4; ++t) {
            v2f b;
            b.x = w0[t * 16];
            b.y = w0[64 + t * 16];
            acc[t] = wmma_k4(a, b, acc[t]);
        }
    }

    int mbase = (lane < 16) ? 0 : 8;
#pragma unroll
    for (int t = 0; t < 4; ++t) {
        int c = t * 16 + mrow;
        float w1024 = Wd1[(size_t)1024 * 64 + c];
        float bc = bd1[c];
#pragma unroll
        for (int j = 0; j < 8; ++j) {
            int mm = mbase + j;
            float v = acc[t][j] + expw[base + mm] * w1024 + bc;
            lds[(wave * 16 + mm) * 64 + c] = fmaxf(v, 0.0f);
        }
    }
    __syncthreads();

    int tid = threadIdx.x;
    if (tid < 128) {
        float s = 0.0f;
#pragma unroll 8
        for (int c = 0; c < 64; ++c) s += lds[tid * 64 + c] * Wd2[c];
        out[blockIdx.x * 128 + tid] = s + bd2[0];
    }
}

extern "C" void kernel_launch(void* const* d_in, const int* in_sizes, int n_in,
                              void* d_out, int out_size, void* d_ws, size_t ws_size,
                              hipStream_t stream) {
    const float* x       = (const float*)d_in[0];
    const float* ew      = (const float*)d_in[1];
    const float* expw    = (const float*)d_in[2];
    const float* W1_rel  = (const float*)d_in[3];
    const float* b1      = (const float*)d_in[4];
    const float* W1_root = (const float*)d_in[5];
    const float* W2_rel  = (const float*)d_in[6];
    const float* b2      = (const float*)d_in[7];
    const float* W2_root = (const float*)d_in[8];
    const float* Wd1     = (const float*)d_in[9];
    const float* bd1     = (const float*)d_in[10];
    const float* Wd2     = (const float*)d_in[11];
    const float* bd2     = (const float*)d_in[12];
    const int*   eidx    = (const int*)d_in[13];
    const int*   eli     = (const int*)d_in[14];

    const int E = in_sizes[1];               // 160000
    const int Nn = in_sizes[0] / FEAT;       // 10000
    const int nfeat = Nn * FEAT;

    float* cnt = (float*)d_ws;
    float* agg = cnt + 16384;                // padded for alignment
    float* z1  = agg + nfeat;
    float* z2  = z1 + nfeat;

    const int* srcI = eidx;
    const int* dstI = eidx + E;

    const int TB = 256;
    dim3 blk(TB);
    long long scatThreads = (long long)E * 128;
    int scatBlocks = (int)((scatThreads + TB - 1) / TB);
    int gemmBlocks = (Nn / 16) * (FEAT / 64) / 8;     // 625
    int decBlocks = E / 128;                          // 1250

    // Layer 1
    zero_kernel<<<(16384 + nfeat + TB - 1) / TB, blk, 0, stream>>>(cnt, 16384 + nfeat);
    count_kernel<<<(E + TB - 1) / TB, blk, 0, stream>>>(dstI, cnt, E);
    scatter_kernel<<<scatBlocks, blk, 0, stream>>>(x, srcI, dstI, ew, agg, E);
    normalize_kernel<<<(nfeat + TB - 1) / TB, blk, 0, stream>>>(agg, cnt, nfeat);
    node_gemm_dual<<<gemmBlocks, blk, 0, stream>>>(agg, W1_rel, x, W1_root, b1, z1, Nn, 1);

    // Layer 2
    zero_kernel<<<(nfeat + TB - 1) / TB, blk, 0, stream>>>(agg, nfeat);
    scatter_kernel<<<scatBlocks, blk, 0, stream>>>(z1, srcI, dstI, ew, agg, E);
    normalize_kernel<<<(nfeat + TB - 1) / TB, blk, 0, stream>>>(agg, cnt, nfeat);
    node_gemm_dual<<<gemmBlocks, blk, 0, stream>>>(agg, W2_rel, z1, W2_root, b2, z2, Nn, 0);

    // Decoder
    decode_kernel<<<decBlocks, blk, 0, stream>>>(z2, eli, expw, Wd1, bd1, Wd2, bd2,
                                                 (float*)d_out, E);
}